// GCN_42915313222045
// MI455X (gfx1250) — compile-verified
//
#include <hip/hip_runtime.h>
#include <hip/hip_bf16.h>
#include <math.h>

// ---------------------------------------------------------------------------
// GCN 2-layer forward for MI455X (gfx1250, wave32).
// GEMMs use V_WMMA_F32_16X16X4_F32 (true fp32 matrix op, matches fp32 ref).
// Message passing uses global float atomics (bandwidth-bound part).
// ---------------------------------------------------------------------------

typedef __attribute__((ext_vector_type(2))) float v2f;
typedef __attribute__((ext_vector_type(8))) float v8f;

#define GCN_N 50000
#define GCN_E 800000

// ---------------- degree / normalization ----------------

__global__ void gcn_deg_kernel(const int* __restrict__ col, float* __restrict__ deg, int e_cnt) {
    int e = blockIdx.x * blockDim.x + threadIdx.x;
    if (e < e_cnt) {
        atomicAdd(&deg[col[e]], 1.0f);
    }
}

__global__ void gcn_dinv_kernel(float* __restrict__ deg, int n) {
    int i = blockIdx.x * blockDim.x + threadIdx.x;
    if (i < n) {
        // +1.0f accounts for the self-loop; always > 0 so no zero guard needed
        deg[i] = rsqrtf(deg[i] + 1.0f);
    }
}

// ---------------- fp32 WMMA GEMM: C[M,NOUT] = A[M,K] @ W[K,NOUT] ----------------
// One wave32 computes one 16x16 output tile via V_WMMA_F32_16X16X4_F32,
// looping K in steps of 4. M (=50000) is an exact multiple of 16 and
// K/NOUT are multiples of 16, so no partial tiles: EXEC is all-ones for
// every wave that reaches the WMMA (ISA requirement).
template <int K, int NOUT>
__global__ void gcn_gemm_wmma(const float* __restrict__ A,
                              const float* __restrict__ W,
                              float* __restrict__ C, int M) {
    const int lane   = threadIdx.x & 31;
    const int waveId = (blockIdx.x * blockDim.x + threadIdx.x) >> 5;
    const int tilesN = NOUT / 16;
    const int m0 = (waveId / tilesN) * 16;
    const int n0 = (waveId % tilesN) * 16;
    if (m0 >= M) return;  // uniform per-wave exit: full waves keep EXEC == ~0

    const int l15   = lane & 15;        // M index (A) / N index (B,D) within tile
    const int khalf = (lane >> 4) * 2;  // lanes 0-15 hold K+{0,1}; lanes 16-31 K+{2,3}

    const float* __restrict__ arow = A + (size_t)(m0 + l15) * K;
    const int n = n0 + l15;

    v8f c = {0.f, 0.f, 0.f, 0.f, 0.f, 0.f, 0.f, 0.f};

    #pragma unroll 4
    for (int k = 0; k < K; k += 4) {
        v2f a, b;
        // A fragment (16x4, ISA 32-bit A layout): lane l -> (M=l&15, K=khalf+{0,1})
        a.x = arow[k + khalf];
        a.y = arow[k + khalf + 1];
        // B fragment (4x16): N striped across lanes, K split across lane halves
        b.x = W[(size_t)(k + khalf) * NOUT + n];
        b.y = W[(size_t)(k + khalf + 1) * NOUT + n];
        c = __builtin_amdgcn_wmma_f32_16x16x4_f32(
                /*neg_a=*/false, a, /*neg_b=*/false, b,
                /*c_mod=*/(short)0, c, /*reuse_a=*/false, /*reuse_b=*/false);
    }

    // D layout: VGPR r -> M = r + 8*(lane>=16), N = lane&15
    const int mbase = m0 + (lane >> 4) * 8;
    #pragma unroll
    for (int r = 0; r < 8; ++r) {
        C[(size_t)(mbase + r) * NOUT + n] = c[r];
    }
}

// ---------------- edge scatter: acc[col] += dinv[row]*dinv[col] * xw[row] --------
// F/2 threads per edge, float2 per thread.
template <int F>
__global__ void gcn_scatter_kernel(const int* __restrict__ row,
                                   const int* __restrict__ col,
                                   const float* __restrict__ dinv,
                                   const float* __restrict__ xw,
                                   float* __restrict__ acc, int e_cnt) {
    const int TPE = F / 2;
    int t = blockIdx.x * blockDim.x + threadIdx.x;
    int e = t / TPE;
    int f = (t - e * TPE) * 2;
    if (e >= e_cnt) return;
    int r = row[e];
    int c = col[e];
    float norm = dinv[r] * dinv[c];
    const float2 v = *(const float2*)(xw + (size_t)r * F + f);
    float* dst = acc + (size_t)c * F + f;
    atomicAdd(dst + 0, norm * v.x);
    atomicAdd(dst + 1, norm * v.y);
}

// ---------------- layer-1 finalize: h = relu(acc + dinv^2 * xw + b) (in place) ---
__global__ void gcn_finalize1_kernel(float* __restrict__ acc,
                                     const float* __restrict__ xw,
                                     const float* __restrict__ dinv,
                                     const float* __restrict__ b, int total) {
    int t = blockIdx.x * blockDim.x + threadIdx.x;
    if (t >= total) return;
    int i = t >> 6;   // node  (F = 64)
    int f = t & 63;   // feature
    float d = dinv[i];
    float v = acc[t] + d * d * xw[t] + b[f];
    acc[t] = fmaxf(v, 0.0f);
}

// ---------------- layer-2 finalize + log_softmax (wave per node, 32 classes) ----
__global__ void gcn_finalize2_lsm_kernel(float* __restrict__ out,     // acc, in place
                                         const float* __restrict__ xw2,
                                         const float* __restrict__ dinv,
                                         const float* __restrict__ b, int n) {
    int wave = (blockIdx.x * blockDim.x + threadIdx.x) >> 5;
    int lane = threadIdx.x & 31;     // wave32: lane == class index
    if (wave >= n) return;
    float d = dinv[wave];
    size_t idx = (size_t)wave * 32 + lane;
    float v = out[idx] + d * d * xw2[idx] + b[lane];

    float m = v;
    #pragma unroll
    for (int off = 16; off > 0; off >>= 1) m = fmaxf(m, __shfl_xor(m, off));
    float ex = expf(v - m);
    float s = ex;
    #pragma unroll
    for (int off = 16; off > 0; off >>= 1) s += __shfl_xor(s, off);
    out[idx] = (v - m) - logf(s);
}

// ---------------------------------------------------------------------------

extern "C" void kernel_launch(void* const* d_in, const int* in_sizes, int n_in,
                              void* d_out, int out_size, void* d_ws, size_t ws_size,
                              hipStream_t stream) {
    const float* x   = (const float*)d_in[0];           // [N,128]
    const int*   ei  = (const int*)d_in[1];             // [2,E] flattened
    const float* W1  = (const float*)d_in[2];           // [128,64]
    const float* b1  = (const float*)d_in[3];           // [64]
    const float* W2  = (const float*)d_in[4];           // [64,32]
    const float* b2  = (const float*)d_in[5];           // [32]
    float* out = (float*)d_out;                         // [N,32]

    const int N = GCN_N;
    const int E = GCN_E;
    const int* row = ei;        // edge_index[0] (source)
    const int* col = ei + E;    // edge_index[1] (target)

    // Workspace layout (floats): dinv | xw1 | acc1(h1) | xw2
    float* ws   = (float*)d_ws;
    float* dinv = ws;                         // N      (50048 slot, 16-aligned)
    float* xw1  = dinv + 50048;               // N*64
    float* acc1 = xw1 + (size_t)N * 64;       // N*64
    float* xw2  = acc1 + (size_t)N * 64;      // N*32

    // 1) degree -> dinv_sqrt
    hipMemsetAsync(dinv, 0, (size_t)N * sizeof(float), stream);
    gcn_deg_kernel<<<(E + 255) / 256, 256, 0, stream>>>(col, dinv, E);
    gcn_dinv_kernel<<<(N + 255) / 256, 256, 0, stream>>>(dinv, N);

    // 2) xw1 = x @ W1   (3125 x 4 tiles of 16x16, one wave each)
    {
        int waves = (N / 16) * (64 / 16);
        int threads = waves * 32;
        gcn_gemm_wmma<128, 64><<<(threads + 255) / 256, 256, 0, stream>>>(x, W1, xw1, N);
    }

    // 3) edge scatter into acc1
    hipMemsetAsync(acc1, 0, (size_t)N * 64 * sizeof(float), stream);
    {
        long long threads = (long long)E * 32;              // 32 threads per edge (F=64)
        gcn_scatter_kernel<64><<<(int)((threads + 255) / 256), 256, 0, stream>>>(
            row, col, dinv, xw1, acc1, E);
    }

    // 4) h1 = relu(acc1 + dinv^2*xw1 + b1), in place
    {
        int total = N * 64;
        gcn_finalize1_kernel<<<(total + 255) / 256, 256, 0, stream>>>(acc1, xw1, dinv, b1, total);
    }

    // 5) xw2 = h1 @ W2
    {
        int waves = (N / 16) * (32 / 16);
        int threads = waves * 32;
        gcn_gemm_wmma<64, 32><<<(threads + 255) / 256, 256, 0, stream>>>(acc1, W2, xw2, N);
    }

    // 6) edge scatter into d_out (used as the layer-2 accumulator)
    hipMemsetAsync(out, 0, (size_t)N * 32 * sizeof(float), stream);
    {
        long long threads = (long long)E * 16;              // 16 threads per edge (F=32)
        gcn_scatter_kernel<32><<<(int)((threads + 255) / 256), 256, 0, stream>>>(
            row, col, dinv, xw2, out, E);
    }

    // 7) self-loop + bias + log_softmax, one wave per node, in place on d_out
    {
        int threads = N * 32;
        gcn_finalize2_lsm_kernel<<<(threads + 255) / 256, 256, 0, stream>>>(out, xw2, dinv, b2, N);
    }

    (void)in_sizes; (void)n_in; (void)out_size; (void)ws_size;
}